// SparseLinear_33251636806222
// MI455X (gfx1250) — compile-verified
//
#include <hip/hip_runtime.h>
#include <cstdint>

// SparseLinear: out[8192,4096] = x[8192,4096] @ (W*mask)^T[4096,4096] + b
// fp32 WMMA (v_wmma_f32_16x16x4_f32) GEMM, mask fused into B-tile load.
// Block tile 128(M) x 256(N), BK=16 double-buffered; 8 waves, each owning a
// 64x64 C tile (16 x v8f accumulators) -> 2:1 wmma:ds_b64 ratio.

#define M_DIM 8192
#define N_DIM 4096
#define K_DIM 4096

#define BM 128
#define BN 256
#define BK 16
#define LDS_STRIDE 20          // BK + 4 pad: multiple of 4 (aligned f4 stores), 20*m mod 64 distinct for m=0..15
#define NSTAGE (K_DIM / BK)    // 256

typedef __attribute__((ext_vector_type(2))) float v2f;
typedef __attribute__((ext_vector_type(8))) float v8f;

__global__ __launch_bounds__(256)
void sparse_linear_wmma_f32(const float* __restrict__ X,
                            const float* __restrict__ W,
                            const uint8_t* __restrict__ Mk,
                            const float* __restrict__ Bias,
                            float* __restrict__ Out)
{
    __shared__ float As[2][BM * LDS_STRIDE];   // 2 x 10 KB
    __shared__ float Bs[2][BN * LDS_STRIDE];   // 2 x 20 KB

    const int tid  = threadIdx.x;
    const int m0   = blockIdx.y * BM;
    const int n0   = blockIdx.x * BN;

    const int wave = tid >> 5;          // 0..7
    const int lane = tid & 31;
    const int wm   = (wave & 1) * 64;   // wave M offset inside block tile
    const int wn   = (wave >> 1) * 64;  // wave N offset inside block tile
    const int lm   = lane & 15;
    const int lhi  = lane >> 4;         // 0 or 1
    const int lk2  = lhi * 2;           // K sub-offset per ISA fp32 A/B layout

    // Global-load mapping:
    //  A tile: 128 rows x 4 float4/row = 512 float4 -> 2 per thread
    //  B tile: 256 rows x 4 float4/row = 1024 float4 -> 4 per thread
    const int ra0 = (tid      ) >> 2, ca0 = ((tid      ) & 3) * 4;
    const int ra1 = (tid + 256) >> 2, ca1 = ((tid + 256) & 3) * 4;
    int rb[4], cb[4];
#pragma unroll
    for (int i = 0; i < 4; ++i) {
        const int idx = tid + i * 256;
        rb[i] = idx >> 2;
        cb[i] = (idx & 3) * 4;
    }

    v8f acc[4][4];
#pragma unroll
    for (int mi = 0; mi < 4; ++mi)
#pragma unroll
        for (int ni = 0; ni < 4; ++ni)
            acc[mi][ni] = (v8f){0.f,0.f,0.f,0.f,0.f,0.f,0.f,0.f};

    float4 a0, a1;
    float4 bw[4];

    auto load_stage = [&](int ks) {
        const size_t kbase = (size_t)ks * BK;
        // A tile: x rows m0..m0+127
        a0 = *reinterpret_cast<const float4*>(X + (size_t)(m0 + ra0) * K_DIM + kbase + ca0);
        a1 = *reinterpret_cast<const float4*>(X + (size_t)(m0 + ra1) * K_DIM + kbase + ca1);
        // B tile: Ws rows n0..n0+255 (mask fused: Ws = mask ? W : 0)
#pragma unroll
        for (int i = 0; i < 4; ++i) {
            const size_t off = (size_t)(n0 + rb[i]) * K_DIM + kbase + cb[i];
            float4 w = *reinterpret_cast<const float4*>(W + off);
            const uint32_t mu = *reinterpret_cast<const uint32_t*>(Mk + off);
            bw[i].x = (mu & 0x000000FFu) ? w.x : 0.f;
            bw[i].y = (mu & 0x0000FF00u) ? w.y : 0.f;
            bw[i].z = (mu & 0x00FF0000u) ? w.z : 0.f;
            bw[i].w = (mu & 0xFF000000u) ? w.w : 0.f;
        }
    };

    auto store_stage = [&](int buf) {
        *reinterpret_cast<float4*>(&As[buf][ra0 * LDS_STRIDE + ca0]) = a0;
        *reinterpret_cast<float4*>(&As[buf][ra1 * LDS_STRIDE + ca1]) = a1;
#pragma unroll
        for (int i = 0; i < 4; ++i)
            *reinterpret_cast<float4*>(&Bs[buf][rb[i] * LDS_STRIDE + cb[i]]) = bw[i];
    };

    auto compute_stage = [&](int buf) {
#pragma unroll
        for (int kk = 0; kk < BK / 4; ++kk) {
            const int kb = kk * 4 + lk2;   // even -> 8B-aligned b64 LDS reads
            v2f afr[4], bfr[4];
#pragma unroll
            for (int mi = 0; mi < 4; ++mi)
                afr[mi] = *reinterpret_cast<const v2f*>(
                    &As[buf][(wm + mi * 16 + lm) * LDS_STRIDE + kb]);
#pragma unroll
            for (int ni = 0; ni < 4; ++ni)
                bfr[ni] = *reinterpret_cast<const v2f*>(
                    &Bs[buf][(wn + ni * 16 + lm) * LDS_STRIDE + kb]);
#pragma unroll
            for (int mi = 0; mi < 4; ++mi)
#pragma unroll
                for (int ni = 0; ni < 4; ++ni)
                    acc[mi][ni] = __builtin_amdgcn_wmma_f32_16x16x4_f32(
                        /*neg_a=*/false, afr[mi],
                        /*neg_b=*/false, bfr[ni],
                        /*c_mod=*/(short)0, acc[mi][ni],
                        /*reuse_a=*/false, /*reuse_b=*/false);
        }
    };

    // Prologue: fill buffer 0
    load_stage(0);
    store_stage(0);
    __syncthreads();

    for (int ks = 0; ks < NSTAGE; ++ks) {
        const int cur = ks & 1;
        if (ks + 1 < NSTAGE) load_stage(ks + 1);   // overlap global loads with math
        compute_stage(cur);
        __syncthreads();                            // everyone done reading LDS buffer cur
        if (ks + 1 < NSTAGE) {
            store_stage(cur ^ 1);
            __syncthreads();
        }
    }

    // Epilogue: C layout (ISA 7.12.2): VGPR r -> row r + 8*(lane>>4), col lane&15
#pragma unroll
    for (int ni = 0; ni < 4; ++ni) {
        const int n = n0 + wn + ni * 16 + lm;
        const float bias = Bias[n];
#pragma unroll
        for (int mi = 0; mi < 4; ++mi) {
#pragma unroll
            for (int r = 0; r < 8; ++r) {
                const int m = m0 + wm + mi * 16 + r + lhi * 8;
                Out[(size_t)m * N_DIM + n] = acc[mi][ni][r] + bias;
            }
        }
    }
}

extern "C" void kernel_launch(void* const* d_in, const int* in_sizes, int n_in,
                              void* d_out, int out_size, void* d_ws, size_t ws_size,
                              hipStream_t stream) {
    const float*   x    = (const float*)d_in[0];
    const float*   W    = (const float*)d_in[1];
    const uint8_t* mask = (const uint8_t*)d_in[2];   // jnp bool -> 1 byte/elem
    const float*   b    = (const float*)d_in[3];
    float*         out  = (float*)d_out;

    dim3 grid(N_DIM / BN, M_DIM / BM);   // 16 x 64 blocks
    dim3 block(256);                     // 8 waves (wave32)
    sparse_linear_wmma_f32<<<grid, block, 0, stream>>>(x, W, mask, b, out);
}